// SubSampling_51513837748790
// MI455X (gfx1250) — compile-verified
//
#include <hip/hip_runtime.h>
#include <stdint.h>

// LeNet subsampling: 2x2 stride-2 SUM pooling + per-channel scale.
// x: [32, 32, 512, 512] f32  ->  out: [32, 32, 256, 256] f32
//
// Pure HBM-streaming problem: ~1.34 GB of traffic, ~0.27 GFLOP.
// Floor at 23.3 TB/s is ~58 us; optimize the memory path only.

typedef __attribute__((ext_vector_type(4))) float v4f;
typedef __attribute__((ext_vector_type(2))) float v2f;

#define IMG_H 512
#define IMG_W 512
#define OUT_H 256
#define OUT_W 256
#define N_IMAGES (32 * 32)   // B*C
#define C_CHANS 32

// Block: 256 threads -> 2 output rows of one (b,c) image (512 outputs, 2/thread).
// c is block-uniform -> coefficient load becomes a scalar load.
__global__ __launch_bounds__(256) void lenet_subsample2x2(
    const float* __restrict__ x,
    const float* __restrict__ coef,
    float* __restrict__ out)
{
    const int tid   = (int)threadIdx.x;
    const int image = (int)(blockIdx.x >> 7);                    // b*32 + c (scalar)
    const int ho    = (((int)blockIdx.x & 127) << 1) | (tid >> 7); // 0..255
    const int p     = tid & 127;                                 // float4 column slot

    const float c = coef[image & (C_CHANS - 1)];                 // scalar s_load

    // Input: rows 2*ho and 2*ho+1, 4 consecutive floats starting at 4*p.
    const size_t in_base = (size_t)image * (IMG_H * IMG_W)
                         + (size_t)(ho << 1) * IMG_W
                         + ((size_t)p << 2);
    const float* src = x + in_base;

    // Streaming reads: input is read exactly once and is 1 GiB >> 192 MB L2,
    // so non-temporal (TH=NT) is the right cache policy.
    v4f a = __builtin_nontemporal_load((const v4f*)src);          // row 2*ho
    v4f b = __builtin_nontemporal_load((const v4f*)(src + IMG_W)); // row 2*ho+1

    v2f r;
    r.x = ((a.x + a.y) + (b.x + b.y)) * c;
    r.y = ((a.z + a.w) + (b.z + b.w)) * c;

    v2f* o = (v2f*)(out + (size_t)image * (OUT_H * OUT_W)
                        + (size_t)ho * OUT_W) + p;
    __builtin_nontemporal_store(r, o);                            // write-once: NT
}

#if defined(__gfx1250__)
// Compile-probe for the CDNA5 async memory path (never launched):
// global_load_async_to_lds_b128 (ASYNCcnt) + s_wait_asynccnt.
// Tells us via the histogram whether this mnemonic assembles on this
// toolchain; a later round can move the streaming loads onto it if useful.
__global__ void cdna5_async_probe(const float* __restrict__ g,
                                  float* __restrict__ out)
{
    __shared__ float tile[256 * 4];
    unsigned lds_addr = (unsigned)(uintptr_t)&tile[threadIdx.x * 4];
    const float* src = g + threadIdx.x * 4;
    asm volatile("global_load_async_to_lds_b128 %0, %1, off"
                 :
                 : "v"(lds_addr), "v"(src)
                 : "memory");
    asm volatile("s_wait_asynccnt 0" ::: "memory");
    __syncthreads();
    out[threadIdx.x] = tile[threadIdx.x];
}
#endif

extern "C" void kernel_launch(void* const* d_in, const int* in_sizes, int n_in,
                              void* d_out, int out_size, void* d_ws, size_t ws_size,
                              hipStream_t stream) {
    const float* x    = (const float*)d_in[0];   // [32,32,512,512] f32
    const float* coef = (const float*)d_in[1];   // [32] f32
    float* out        = (float*)d_out;           // [32,32,256,256] f32

    // 1024 images * 128 row-pairs per image
    dim3 grid(N_IMAGES * (OUT_H / 2));
    dim3 block(256);
    lenet_subsample2x2<<<grid, block, 0, stream>>>(x, coef, out);
}